// Morton_encode_69312182223577
// MI455X (gfx1250) — compile-verified
//
#include <hip/hip_runtime.h>
#include <hip/hip_bf16.h>
#include <stdint.h>

// Morton (Z-order) permutation of (B, C, 256, 256) f32:
//   out[bc, m] = x[bc, i, j],  m = interleave(i, j)  (even bits = j, odd = i)
//
// Strategy (bandwidth-bound):
//   - 1 workgroup (256 thr) per 64x64 input tile; its Morton codes are 4096
//     consecutive output indices.
//   - Phase 1: CDNA5 async DMA  global_load_async_to_lds_b128 (ASYNCcnt path),
//     coalesced 128b reads, XOR-swizzled LDS layout.
//   - Phase 2: conflict-free ds_load_b64 pairs -> coalesced 128b NT stores.

typedef float v2f __attribute__((ext_vector_type(2)));
typedef float v4f __attribute__((ext_vector_type(4)));

__device__ __forceinline__ uint32_t compact_even_bits(uint32_t x) {
    // Gather bits 0,2,4,... of x into a dense low field.
    x &= 0x55555555u;
    x = (x | (x >> 1)) & 0x33333333u;
    x = (x | (x >> 2)) & 0x0F0F0F0Fu;
    x = (x | (x >> 4)) & 0x00FF00FFu;
    x = (x | (x >> 8)) & 0x0000FFFFu;
    return x;
}

__global__ __launch_bounds__(256) void morton_tile_kernel(
    const float* __restrict__ x, float* __restrict__ out) {
    __shared__ float lds[64 * 64];  // 16 KB tile

    const uint32_t tid = threadIdx.x;
    const uint32_t w   = blockIdx.x;
    const uint32_t bc  = w >> 4;      // which 256x256 image
    const uint32_t t   = w & 15u;     // Morton-ordered 64x64 tile within image
    // t = interleave(ti, tj): t0=tj0, t1=ti0, t2=tj1, t3=ti1
    const uint32_t tj = (t & 1u) | ((t >> 1) & 2u);
    const uint32_t ti = ((t >> 1) & 1u) | ((t >> 2) & 2u);

    // ---------------- Phase 1: async copy 64x64 tile into LDS ----------------
    // Source rows: x[bc, ti*64 + il, tj*64 + 0..63], row stride 256 floats.
    const float* src = x + ((size_t)bc << 16) + ((size_t)ti << 14) + ((size_t)tj << 6);
    #pragma unroll
    for (int k = 0; k < 4; ++k) {
        const uint32_t id = tid + (uint32_t)k * 256u;  // 0..1023 b128 chunks
        const uint32_t il = id >> 4;                   // row in tile (0..63)
        const uint32_t jl = (id & 15u) << 2;           // col (multiple of 4)
        const uint32_t sw = ((il >> 1) & 3u) << 4;     // XOR swizzle, bits 4..5
        const uint32_t ldsOff =
            (uint32_t)(size_t)&lds[il * 64u + (jl ^ sw)];      // LDS byte addr
        const uint64_t gaddr = (uint64_t)(size_t)(src + il * 256u + jl);
        asm volatile("global_load_async_to_lds_b128 %0, %1, off"
                     :
                     : "v"(ldsOff), "v"(gaddr)
                     : "memory");
    }
    asm volatile("s_wait_asynccnt 0x0" ::: "memory");  // my wave's DMA landed
    __syncthreads();                                   // everyone's DMA landed

    // ---------------- Phase 2: Morton-ordered coalesced stores ---------------
    // Output block for this tile: out[bc*65536 + t*4096 + 0..4095].
    float* dst = out + ((size_t)bc << 16) + ((size_t)t << 12);
    #pragma unroll
    for (int k = 0; k < 4; ++k) {
        const uint32_t q  = tid + (uint32_t)k * 256u;  // float4 index (0..1023)
        // m = 4q + r, r in 0..3 covers the 2x2 block at (il, jl):
        const uint32_t jl = compact_even_bits(q) << 1;        // even
        const uint32_t il = compact_even_bits(q >> 1) << 1;   // even
        const uint32_t sw  = ((il >> 1) & 3u) << 4;
        const uint32_t off = il * 64u + (jl ^ sw);
        // conflict-free across the wave: banks = distinct even pairs
        const v2f r0 = *(const v2f*)&lds[off];        // row il
        const v2f r1 = *(const v2f*)&lds[off + 64u];  // row il+1 (same sw)
        v4f v;
        v.x = r0.x;  // m=4q+0 -> (il,   jl)
        v.y = r0.y;  // m=4q+1 -> (il,   jl+1)
        v.z = r1.x;  // m=4q+2 -> (il+1, jl)
        v.w = r1.y;  // m=4q+3 -> (il+1, jl+1)
        __builtin_nontemporal_store(v, (v4f*)(dst + ((size_t)q << 2)));
    }
}

extern "C" void kernel_launch(void* const* d_in, const int* in_sizes, int n_in,
                              void* d_out, int out_size, void* d_ws, size_t ws_size,
                              hipStream_t stream) {
    (void)n_in; (void)out_size; (void)d_ws; (void)ws_size;
    const float* x  = (const float*)d_in[0];
    float* out      = (float*)d_out;
    const int total   = in_sizes[0];        // B*C*256*256
    const int nImages = total >> 16;        // B*C (65536 elems per image)
    const int nBlocks = nImages * 16;       // 16 Morton tiles of 64x64 per image
    morton_tile_kernel<<<nBlocks, 256, 0, stream>>>(x, out);
}